// GraphSAGE_35828617183789
// MI455X (gfx1250) — compile-verified
//
#include <hip/hip_runtime.h>
#include <hip/hip_bf16.h>

typedef __attribute__((ext_vector_type(2))) float v2f;
typedef __attribute__((ext_vector_type(8))) float v8f;

#define IN_DIM  64
#define HID_DIM 128
#define OUT_DIM 2
#define BN_EPS  1e-5f

// ---------------------------------------------------------------------------
// zero a flat float region
__global__ void zero_kernel(float* __restrict__ p, long long count) {
    long long tid = (long long)blockIdx.x * blockDim.x + threadIdx.x;
    if (tid < count) p[tid] = 0.0f;
}

// ---------------------------------------------------------------------------
// Layer-1 scatter: agg1[dst] += x[src] (per feature), cnt[dst] += 1
__global__ void scatter1_kernel(const float* __restrict__ x,
                                const int* __restrict__ src,
                                const int* __restrict__ dst,
                                float* __restrict__ agg1,
                                float* __restrict__ cnt,
                                long long total) {
    long long tid = (long long)blockIdx.x * blockDim.x + threadIdx.x;
    if (tid >= total) return;
    int e = (int)(tid >> 6);       // edge id
    int t = (int)(tid & 63);       // feature id
    int s = src[e];
    int d = dst[e];
    unsafeAtomicAdd(&agg1[(long long)d * IN_DIM + t], x[(long long)s * IN_DIM + t]);
    if (t == 0) unsafeAtomicAdd(&cnt[d], 1.0f);
}

// ---------------------------------------------------------------------------
// Layer-1 GEMM via V_WMMA_F32_16X16X4_F32:
//   h = (agg1/cnt) @ W1l + x @ W1r + b1      (h is N x 128)
// plus fused BatchNorm statistics (column sums / sums of squares).
// One wave computes one 16x16 tile; 8 waves/block cover all 128 columns.
__global__ __launch_bounds__(256) void gemm1_kernel(
    const float* __restrict__ x, const float* __restrict__ agg1,
    const float* __restrict__ cnt,
    const float* __restrict__ W1l, const float* __restrict__ W1r,
    const float* __restrict__ b1,
    float* __restrict__ h, float* __restrict__ colsum, float* __restrict__ colsq) {

    __shared__ float ssum[HID_DIM];
    __shared__ float ssq[HID_DIM];
    if (threadIdx.x < HID_DIM) { ssum[threadIdx.x] = 0.0f; ssq[threadIdx.x] = 0.0f; }
    __syncthreads();

    const int wave = threadIdx.x >> 5;
    const int lane = threadIdx.x & 31;
    const int lr   = lane & 15;
    const int hi   = lane >> 4;

    const int row_base = blockIdx.x * 16;        // N divisible by 16
    const int col_base = wave * 16;
    const int col      = col_base + lr;

    const int arow  = row_base + lr;             // A-fragment row for this lane
    const float inv = 1.0f / fmaxf(cnt[arow], 1.0f);

    v8f c;
    const float bias = b1[col];
#pragma unroll
    for (int i = 0; i < 8; ++i) c[i] = bias;

    // term 1: (agg1 / cnt) @ W1l   (K = 64)
#pragma unroll
    for (int kk = 0; kk < IN_DIM; kk += 4) {
        const int k0 = kk + hi * 2;
        v2f a, b;
        a.x = agg1[arow * IN_DIM + k0]     * inv;
        a.y = agg1[arow * IN_DIM + k0 + 1] * inv;
        b.x = W1l[(k0)     * HID_DIM + col];
        b.y = W1l[(k0 + 1) * HID_DIM + col];
        c = __builtin_amdgcn_wmma_f32_16x16x4_f32(false, a, false, b, (short)0, c, false, false);
    }
    // term 2: x @ W1r   (K = 64)
#pragma unroll
    for (int kk = 0; kk < IN_DIM; kk += 4) {
        const int k0 = kk + hi * 2;
        v2f a, b;
        a.x = x[arow * IN_DIM + k0];
        a.y = x[arow * IN_DIM + k0 + 1];
        b.x = W1r[(k0)     * HID_DIM + col];
        b.y = W1r[(k0 + 1) * HID_DIM + col];
        c = __builtin_amdgcn_wmma_f32_16x16x4_f32(false, a, false, b, (short)0, c, false, false);
    }

    // store h tile + BN partial sums (each lane owns a single column)
    float s = 0.0f, s2 = 0.0f;
#pragma unroll
    for (int i = 0; i < 8; ++i) {
        const int rout = row_base + hi * 8 + i;
        h[(long long)rout * HID_DIM + col] = c[i];
        s  += c[i];
        s2 += c[i] * c[i];
    }
    unsafeAtomicAdd(&ssum[col], s);
    unsafeAtomicAdd(&ssq[col], s2);
    __syncthreads();
    if (threadIdx.x < HID_DIM) {
        unsafeAtomicAdd(&colsum[threadIdx.x], ssum[threadIdx.x]);
        unsafeAtomicAdd(&colsq[threadIdx.x], ssq[threadIdx.x]);
    }
}

// ---------------------------------------------------------------------------
// Fold BatchNorm into per-column scale/shift: hn = h*scale + shift, then ReLU.
__global__ void bn_kernel(const float* __restrict__ colsum, const float* __restrict__ colsq,
                          const float* __restrict__ gamma, const float* __restrict__ beta,
                          float* __restrict__ scale, float* __restrict__ shift, float invN) {
    int t = threadIdx.x;
    float mu  = colsum[t] * invN;
    float var = colsq[t] * invN - mu * mu;
    float rs  = rsqrtf(var + BN_EPS);
    float g   = rs * gamma[t];
    scale[t] = g;
    shift[t] = beta[t] - mu * g;
}

// ---------------------------------------------------------------------------
// Layer-2 scatter: agg2[dst] += relu(bn(h[src]))   (normalization applied on the fly)
__global__ void scatter2_kernel(const float* __restrict__ h,
                                const int* __restrict__ src,
                                const int* __restrict__ dst,
                                const float* __restrict__ scale,
                                const float* __restrict__ shift,
                                float* __restrict__ agg2,
                                long long total) {
    long long tid = (long long)blockIdx.x * blockDim.x + threadIdx.x;
    if (tid >= total) return;
    int e = (int)(tid >> 7);
    int t = (int)(tid & 127);
    int s = src[e];
    int d = dst[e];
    float v  = h[(long long)s * HID_DIM + t];
    float hn = fmaxf(v * scale[t] + shift[t], 0.0f);
    unsafeAtomicAdd(&agg2[(long long)d * HID_DIM + t], hn);
}

// ---------------------------------------------------------------------------
// Layer-2 output: out = (agg2/cnt) @ W2l + relu(bn(h)) @ W2r + b2  (OUT_DIM=2)
// One wave32 per node, shuffle reduction.
__global__ __launch_bounds__(256) void layer2_out_kernel(
    const float* __restrict__ h, const float* __restrict__ agg2,
    const float* __restrict__ cnt,
    const float* __restrict__ scale, const float* __restrict__ shift,
    const float* __restrict__ W2l, const float* __restrict__ W2r,
    const float* __restrict__ b2, float* __restrict__ out, int n) {
    int wave = threadIdx.x >> 5;
    int lane = threadIdx.x & 31;
    int node = blockIdx.x * 8 + wave;
    if (node >= n) return;
    float inv = 1.0f / fmaxf(cnt[node], 1.0f);
    float acc0 = 0.0f, acc1 = 0.0f;
#pragma unroll
    for (int j = 0; j < 4; ++j) {
        int k = lane + j * 32;
        float a2 = agg2[(long long)node * HID_DIM + k] * inv;
        float hk = fmaxf(h[(long long)node * HID_DIM + k] * scale[k] + shift[k], 0.0f);
        acc0 += a2 * W2l[k * OUT_DIM + 0] + hk * W2r[k * OUT_DIM + 0];
        acc1 += a2 * W2l[k * OUT_DIM + 1] + hk * W2r[k * OUT_DIM + 1];
    }
#pragma unroll
    for (int off = 16; off > 0; off >>= 1) {
        acc0 += __shfl_down(acc0, off, 32);
        acc1 += __shfl_down(acc1, off, 32);
    }
    if (lane == 0) {
        out[node * OUT_DIM + 0] = acc0 + b2[0];
        out[node * OUT_DIM + 1] = acc1 + b2[1];
    }
}

// ---------------------------------------------------------------------------
extern "C" void kernel_launch(void* const* d_in, const int* in_sizes, int n_in,
                              void* d_out, int out_size, void* d_ws, size_t ws_size,
                              hipStream_t stream) {
    const float* x     = (const float*)d_in[0];
    const int*   edges = (const int*)d_in[1];          // (2, E) flat: src then dst
    const float* W1l   = (const float*)d_in[2];
    const float* W1r   = (const float*)d_in[3];
    const float* b1    = (const float*)d_in[4];
    const float* gamma = (const float*)d_in[5];
    const float* beta  = (const float*)d_in[6];
    const float* W2l   = (const float*)d_in[7];
    const float* W2r   = (const float*)d_in[8];
    const float* b2    = (const float*)d_in[9];
    float* out = (float*)d_out;

    const int N = in_sizes[0] / IN_DIM;   // 100000
    const int E = in_sizes[1] / 2;        // 1000000
    const int* src = edges;
    const int* dst = edges + E;

    // workspace layout (floats); agg1 (N*64) overlays agg2 (N*128): agg1 dead
    // after gemm1, agg2 zeroed right before scatter2.
    float* ws = (float*)d_ws;
    float* h      = ws;                                   // N*128
    float* aggRgn = ws + (size_t)N * HID_DIM;             // N*128 (agg1/agg2 overlay)
    float* cnt    = aggRgn + (size_t)N * HID_DIM;         // N
    float* colsum = cnt + N;                              // 128
    float* colsq  = colsum + HID_DIM;                     // 128
    float* scale  = colsq + HID_DIM;                      // 128
    float* shift  = scale + HID_DIM;                      // 128

    const int TB = 256;

    // zero agg region + cnt + BN accumulators (covers agg1 and later agg2 base)
    {
        long long cnt0 = (long long)N * HID_DIM + N + 4 * HID_DIM;
        int grid = (int)((cnt0 + TB - 1) / TB);
        zero_kernel<<<grid, TB, 0, stream>>>(aggRgn, cnt0);
    }
    // scatter layer 1
    {
        long long total = (long long)E * IN_DIM;
        int grid = (int)((total + TB - 1) / TB);
        scatter1_kernel<<<grid, TB, 0, stream>>>(x, src, dst, aggRgn, cnt, total);
    }
    // fused GEMM1 + bias + BN stats (WMMA f32 16x16x4)
    {
        int grid = N / 16;   // 6250, N divisible by 16
        gemm1_kernel<<<grid, TB, 0, stream>>>(x, aggRgn, cnt, W1l, W1r, b1,
                                              h, colsum, colsq);
    }
    // fold BN into scale/shift
    bn_kernel<<<1, HID_DIM, 0, stream>>>(colsum, colsq, gamma, beta, scale, shift,
                                         1.0f / (float)N);
    // re-zero overlay region as agg2
    {
        long long cnt2 = (long long)N * HID_DIM;
        int grid = (int)((cnt2 + TB - 1) / TB);
        zero_kernel<<<grid, TB, 0, stream>>>(aggRgn, cnt2);
    }
    // scatter layer 2 with fused BN+ReLU
    {
        long long total = (long long)E * HID_DIM;
        int grid = (int)((total + TB - 1) / TB);
        scatter2_kernel<<<grid, TB, 0, stream>>>(h, src, dst, scale, shift, aggRgn, total);
    }
    // final projection to OUT_DIM=2
    {
        int grid = (N + 7) / 8;
        layer2_out_kernel<<<grid, TB, 0, stream>>>(h, aggRgn, cnt, scale, shift,
                                                   W2l, W2r, b2, out, N);
    }
}